// Coarse_module_67345087201829
// MI455X (gfx1250) — compile-verified
//
#include <hip/hip_runtime.h>
#include <math.h>

// Problem constants (match reference)
#define NN   500          // nodes
#define TT   12           // timestamps
#define FF   3            // features
#define DIMR (NN*TT)      // 6000
#define ADJ_ELEMS (NN*NN) // 250000
#define OUT_VEC4 (DIMR*(DIMR/4))   // 6000*1500 = 9,000,000 float4 stores

typedef __attribute__((ext_vector_type(2))) float v2f;
typedef __attribute__((ext_vector_type(8))) float v8f;

// ---------------------------------------------------------------------------
// 1) zero the scalar accumulator
__global__ void zero_kernel(float* w) { if (threadIdx.x == 0) w[0] = 0.0f; }

// ---------------------------------------------------------------------------
// 2) sum(adj). Entries are 0.0/1.0 so every partial sum is an exact small
//    integer in f32 -> atomicAdd order does not change the value.
__global__ void reduce_adj(const float* __restrict__ x, int n, float* __restrict__ out) {
    __shared__ float s[256];
    float acc = 0.0f;
    for (int idx = blockIdx.x * 256 + threadIdx.x; idx < n; idx += gridDim.x * 256)
        acc += x[idx];
    s[threadIdx.x] = acc;
    __syncthreads();
    for (int off = 128; off > 0; off >>= 1) {
        if (threadIdx.x < off) s[threadIdx.x] += s[threadIdx.x + off];
        __syncthreads();
    }
    if (threadIdx.x == 0) atomicAdd(out, s[0]);
}

// ---------------------------------------------------------------------------
// 3) infor_IT[t] = A @ F_t  via V_WMMA_F32_16X16X4_F32 (f32 in / f32 acc).
//    One wave32 per (t, 16-row M tile). K = 500 = 125 * 4 exactly.
//    A-frag layout (16x4 f32): lanes 0-15: v0=K0, v1=K1 ; lanes 16-31: v0=K2, v1=K3
//    B-frag layout (4x16 f32): v0: K0 (lanes 0-15) / K2 (lanes 16-31); v1: K1 / K3
//    D layout (16x16 f32)   : vgpr v -> row v (lanes 0-15) / row v+8 (lanes 16-31)
__global__ __launch_bounds__(32) void infor_wmma_kernel(
        const float* __restrict__ adj,      // (NN, NN) row-major
        const float* __restrict__ feat,     // (TT*NN, FF)
        float* __restrict__ inforIT) {      // (TT*NN, FF)
    const int wave  = blockIdx.x;           // 0 .. TT*32-1
    const int t     = wave >> 5;            // /32  (32 M-tiles of 16 cover 500 rows)
    const int mtile = wave & 31;
    const int m0    = mtile * 16;
    const int lane  = threadIdx.x;
    const int half  = lane >> 4;            // 0 or 1
    const int lh    = lane & 15;

    const int rowA   = m0 + lh;
    const bool rowOK = rowA < NN;
    const int rowSafe = rowOK ? rowA : (NN - 1);

    const float* featT = feat + (size_t)t * NN * FF;
    const int nSafe = (lh < FF) ? lh : 0;   // clamp column index for safe loads

    v8f c = {0.f, 0.f, 0.f, 0.f, 0.f, 0.f, 0.f, 0.f};

    for (int k0 = 0; k0 < NN; k0 += 4) {
        const int ka = k0 + 2 * half;       // K index for this half-wave
        // A fragment (clamped row, then masked to 0 for rows >= NN)
        const float* pa = adj + (size_t)rowSafe * NN + ka;
        v2f a;
        a.x = rowOK ? pa[0] : 0.0f;
        a.y = rowOK ? pa[1] : 0.0f;
        // B fragment (features column n for n<FF, else 0)
        float bx = featT[(size_t)ka * FF + nSafe];
        float by = featT[(size_t)(ka + 1) * FF + nSafe];
        v2f b;
        b.x = (lh < FF) ? bx : 0.0f;
        b.y = (lh < FF) ? by : 0.0f;

        c = __builtin_amdgcn_wmma_f32_16x16x4_f32(
                /*neg_a=*/false, a, /*neg_b=*/false, b,
                /*c_mod=*/(short)0, c, /*reuse_a=*/false, /*reuse_b=*/false);
    }

    // Scatter D: only columns 0..FF-1 are meaningful.
    if (lh < FF) {
#pragma unroll
        for (int v = 0; v < 8; ++v) {
            const int rowD = m0 + v + 8 * half;
            if (rowD < NN)
                inforIT[((size_t)t * NN + rowD) * FF + lh] = c[v];
        }
    }
}

// ---------------------------------------------------------------------------
// 4) Per-node gate computation. Folds both exclusive time-prefix sums:
//    infor_CS[t] = cumF (prefix of features), infor_CT[t] = cumG (prefix of infor_IT).
__global__ void rho_kernel(
        const float* __restrict__ feat,      // (TT*NN, FF)
        const float* __restrict__ interv,    // (TT, NN)
        const float* __restrict__ sumAp,     // scalar
        const float* __restrict__ inforIT,   // (TT*NN, FF)
        const float* __restrict__ w1IT, const float* __restrict__ w2IT,
        const float* __restrict__ gwIT, const float* __restrict__ gbIT,
        const float* __restrict__ w1CS, const float* __restrict__ w2CS,
        const float* __restrict__ gwCS, const float* __restrict__ gbCS,
        const float* __restrict__ w1CT, const float* __restrict__ w2CT,
        const float* __restrict__ gwCT, const float* __restrict__ gbCT,
        float* __restrict__ rhoIT, float* __restrict__ rhoCS, float* __restrict__ rhoCT) {
    const int i = blockIdx.x * blockDim.x + threadIdx.x;
    if (i >= NN) return;

    const float sumA  = sumAp[0];
    const float judge = sumA * (float)TT;                 // exact integer math in f32
    const bool fIT = ((float)TT * sumA) > judge;          // equal -> false
    const bool fCS = ((float)(TT * (TT - 1) / 2) * (float)NN) > judge;
    const bool fCT = ((float)(TT * (TT - 1) / 2) * sumA) > judge;

    const float a1IT = w1IT[0], a2IT = w2IT[0], bIT = gbIT[0];
    const float g0IT = gwIT[0], g1IT = gwIT[1], g2IT = gwIT[2];
    const float a1CS = w1CS[0], a2CS = w2CS[0], bCS = gbCS[0];
    const float g0CS = gwCS[0], g1CS = gwCS[1], g2CS = gwCS[2];
    const float a1CT = w1CT[0], a2CT = w2CT[0], bCT = gbCT[0];
    const float g0CT = gwCT[0], g1CT = gwCT[1], g2CT = gwCT[2];

    float cumF0 = 0.f, cumF1 = 0.f, cumF2 = 0.f;   // prefix of features
    float cumG0 = 0.f, cumG1 = 0.f, cumG2 = 0.f;   // prefix of infor_IT
    float cumI  = 0.f;                              // prefix of interven

    for (int t = 0; t < TT; ++t) {
        const int r = t * NN + i;
        const float f0 = feat[(size_t)r * FF + 0];
        const float f1 = feat[(size_t)r * FF + 1];
        const float f2 = feat[(size_t)r * FF + 2];
        const float q0 = inforIT[(size_t)r * FF + 0];
        const float q1 = inforIT[(size_t)r * FF + 1];
        const float q2 = inforIT[(size_t)r * FF + 2];
        const float iv = interv[r];

        const float iaIT = fIT ? cumI : iv;
        const float iaCS = fCS ? cumI : iv;
        const float iaCT = fCT ? cumI : iv;

        float z;
        z = bIT + g0IT * (a1IT * q0 + iaIT + a2IT * f0)
                + g1IT * (a1IT * q1 + iaIT + a2IT * f1)
                + g2IT * (a1IT * q2 + iaIT + a2IT * f2);
        rhoIT[r] = 1.0f / (1.0f + expf(-z));

        z = bCS + g0CS * (a1CS * cumF0 + iaCS + a2CS * f0)
                + g1CS * (a1CS * cumF1 + iaCS + a2CS * f1)
                + g2CS * (a1CS * cumF2 + iaCS + a2CS * f2);
        rhoCS[r] = 1.0f / (1.0f + expf(-z));

        z = bCT + g0CT * (a1CT * cumG0 + iaCT + a2CT * f0)
                + g1CT * (a1CT * cumG1 + iaCT + a2CT * f1)
                + g2CT * (a1CT * cumG2 + iaCT + a2CT * f2);
        rhoCT[r] = 1.0f / (1.0f + expf(-z));

        cumF0 += f0; cumF1 += f1; cumF2 += f2;
        cumG0 += q0; cumG1 += q1; cumG2 += q2;
        cumI  += iv;
    }
}

// ---------------------------------------------------------------------------
// 5) Streaming writer: 144 MB of float4 stores — the bandwidth-bound stage.
//    Out[(t,i),(t',j)] = (t'==t) rhoIT[r] A[i,j]
//                      + (t'<t) (rhoCT[r] A[i,j] + rhoCS[r] [i==j]); 0 above.
//    500 % 4 == 0 so every float4 lies inside one (t',j) block, 16B aligned.
__global__ void writer_kernel(
        const float* __restrict__ adj,
        const float* __restrict__ rhoIT,
        const float* __restrict__ rhoCS,
        const float* __restrict__ rhoCT,
        float* __restrict__ out) {
    const unsigned g = blockIdx.x * 256u + threadIdx.x;
    if (g >= (unsigned)OUT_VEC4) return;
    const unsigned r  = g / 1500u;              // output row (DIMR/4 = 1500 vec4/row)
    const unsigned c4 = (g - r * 1500u) << 2;   // starting column
    const unsigned t  = r / (unsigned)NN;
    const unsigned i  = r - t * (unsigned)NN;
    const unsigned tp = c4 / (unsigned)NN;
    const unsigned j0 = c4 - tp * (unsigned)NN;

    float4 o = make_float4(0.f, 0.f, 0.f, 0.f);
    if (tp <= t) {
        const float4 a4 = *reinterpret_cast<const float4*>(adj + (size_t)i * NN + j0);
        if (tp == t) {
            const float s = rhoIT[r];
            o = make_float4(a4.x * s, a4.y * s, a4.z * s, a4.w * s);
        } else {
            const float s = rhoCT[r];
            const float d = rhoCS[r];
            o = make_float4(a4.x * s + ((j0 + 0u == i) ? d : 0.f),
                            a4.y * s + ((j0 + 1u == i) ? d : 0.f),
                            a4.z * s + ((j0 + 2u == i) ? d : 0.f),
                            a4.w * s + ((j0 + 3u == i) ? d : 0.f));
        }
    }
    reinterpret_cast<float4*>(out)[g] = o;
}

// ---------------------------------------------------------------------------
extern "C" void kernel_launch(void* const* d_in, const int* in_sizes, int n_in,
                              void* d_out, int out_size, void* d_ws, size_t ws_size,
                              hipStream_t stream) {
    const float* feat   = (const float*)d_in[0];   // (T,N,F)
    const float* interv = (const float*)d_in[1];   // (T,N)
    const float* adj    = (const float*)d_in[2];   // (N,N)
    const float* w1IT = (const float*)d_in[3];
    const float* w2IT = (const float*)d_in[4];
    const float* gwIT = (const float*)d_in[5];
    const float* gbIT = (const float*)d_in[6];
    const float* w1CS = (const float*)d_in[7];
    const float* w2CS = (const float*)d_in[8];
    const float* gwCS = (const float*)d_in[9];
    const float* gbCS = (const float*)d_in[10];
    const float* w1CT = (const float*)d_in[11];
    const float* w2CT = (const float*)d_in[12];
    const float* gwCT = (const float*)d_in[13];
    const float* gbCT = (const float*)d_in[14];
    float* out = (float*)d_out;

    // Workspace layout (floats): [0]=sumA, [16..]=inforIT(18000), then 3x rho(6000)
    float* W        = (float*)d_ws;
    float* sumA     = W;
    float* inforIT  = W + 16;
    float* rhoIT    = inforIT + (size_t)TT * NN * FF;
    float* rhoCS    = rhoIT + DIMR;
    float* rhoCT    = rhoCS + DIMR;

    zero_kernel<<<1, 32, 0, stream>>>(sumA);
    reduce_adj<<<64, 256, 0, stream>>>(adj, ADJ_ELEMS, sumA);
    infor_wmma_kernel<<<TT * 32, 32, 0, stream>>>(adj, feat, inforIT);
    rho_kernel<<<(NN + 255) / 256, 256, 0, stream>>>(
        feat, interv, sumA, inforIT,
        w1IT, w2IT, gwIT, gbIT,
        w1CS, w2CS, gwCS, gbCS,
        w1CT, w2CT, gwCT, gbCT,
        rhoIT, rhoCS, rhoCT);
    writer_kernel<<<(OUT_VEC4 + 255) / 256, 256, 0, stream>>>(
        adj, rhoIT, rhoCS, rhoCT, out);
}